// GCN_52656299049171
// MI455X (gfx1250) — compile-verified
//
#include <hip/hip_runtime.h>

typedef float v2f __attribute__((ext_vector_type(2)));
typedef float v4f __attribute__((ext_vector_type(4)));
typedef float v8f __attribute__((ext_vector_type(8)));

#define N_NODES 100000
#define N_EDGES 1600000
#define IN_DIM  165
#define HID     64
#define OUT_DIM 2
#define NTILES  (N_NODES / 16)   // 6250, exact
#define KPAIRS  83               // ceil(166/2): K rows 0..164 + zero pad row 165

// ---------------- utility ----------------
__global__ void zero_kernel(float* __restrict__ p, int n) {
    int i = blockIdx.x * blockDim.x + threadIdx.x;
    if (i < n) p[i] = 0.0f;
}

// deg[dst] += 1 for every edge (deg buffer pre-zeroed)
__global__ void deg_kernel(const int* __restrict__ dst, float* __restrict__ deg) {
    int e = blockIdx.x * blockDim.x + threadIdx.x;
    if (e < N_EDGES) unsafeAtomicAdd(&deg[dst[e]], 1.0f);
}

// in-place: dinv = rsqrt(deg + 1)
__global__ void dinv_kernel(float* __restrict__ d) {
    int i = blockIdx.x * blockDim.x + threadIdx.x;
    if (i < N_NODES) d[i] = rsqrtf(d[i] + 1.0f);
}

__global__ void enorm_kernel(const int* __restrict__ src, const int* __restrict__ dst,
                             const float* __restrict__ dinv, float* __restrict__ en) {
    int e = blockIdx.x * blockDim.x + threadIdx.x;
    if (e < N_EDGES) en[e] = dinv[src[e]] * dinv[dst[e]];
}

// ---------------- GEMM1: hw1[N,64] = x[N,165] @ W1[165,64] via f32 WMMA ----------------
// LDS holds W1 in K-pair-interleaved layout:
//   w1p[kp*128 + col*2 + s] = W1[(2*kp + s)*HID + col]   (0 if K row >= 165)
// so a B fragment {W1[ka][col], W1[ka+1][col]} is ONE aligned ds_load_b64.
__global__ __launch_bounds__(256) void gemm1_wmma(const float* __restrict__ x,
                                                  const float* __restrict__ W1,
                                                  float* __restrict__ hw1) {
    __shared__ float w1p[KPAIRS * 2 * HID];       // 83*128 floats = 42496 B LDS
    for (int i = threadIdx.x; i < KPAIRS * 2 * HID; i += 256) {
        int kp   = i >> 7;            // pair index
        int r    = i & 127;
        int col  = r >> 1;
        int sub  = r & 1;
        int krow = kp * 2 + sub;
        w1p[i] = (krow < IN_DIM) ? W1[krow * HID + col] : 0.0f;
    }
    __syncthreads();

    const int wave = threadIdx.x >> 5;
    const int lane = threadIdx.x & 31;
    const int tile = blockIdx.x * 8 + wave;       // 16-row tile of output
    if (tile >= NTILES) return;                   // wave-uniform exit

    const int hh  = lane >> 4;                    // half-wave selector
    const int lr  = lane & 15;
    const int row = tile * 16 + lr;               // A-matrix row for this lane
    const float* __restrict__ xrb = x + (long)row * IN_DIM;   // row base
    const float* __restrict__ xr  = xrb + 2 * hh;             // lane-adjusted

    v8f acc0 = (v8f){}, acc1 = (v8f){}, acc2 = (v8f){}, acc3 = (v8f){};

    // ---- main loop: K = 0..163, 41 iterations, fully in-bounds, branch-free ----
    for (int k = 0; k < 164; k += 4) {
        v2f A;
        A.x = xr[k];
        A.y = xr[k + 1];
        const int ka = k + 2 * hh;                // even
        const v2f* __restrict__ bp = (const v2f*)&w1p[(ka >> 1) * 2 * HID + lr * 2];
        v2f B0 = bp[0];                           // col = lr
        v2f B1 = bp[16];                          // col = lr + 16
        v2f B2 = bp[32];                          // col = lr + 32
        v2f B3 = bp[48];                          // col = lr + 48
        acc0 = __builtin_amdgcn_wmma_f32_16x16x4_f32(false, A, false, B0, (short)0, acc0, false, false);
        acc1 = __builtin_amdgcn_wmma_f32_16x16x4_f32(false, A, false, B1, (short)0, acc1, false, false);
        acc2 = __builtin_amdgcn_wmma_f32_16x16x4_f32(false, A, false, B2, (short)0, acc2, false, false);
        acc3 = __builtin_amdgcn_wmma_f32_16x16x4_f32(false, A, false, B3, (short)0, acc3, false, false);
    }

    // ---- tail: only K = 164 valid. A's invalid K-slots (165/166/167) are zeroed;
    //      B for hh=1 lanes re-reads the (valid) K=164/165 pair and is multiplied
    //      by A = 0, so no masking needed on the B side. ----
    {
        float a164 = xrb[164];                    // always in-bounds load
        v2f A;
        A.x = (hh == 0) ? a164 : 0.0f;
        A.y = 0.0f;
        const v2f* __restrict__ bp = (const v2f*)&w1p[(KPAIRS - 1) * 2 * HID + lr * 2];
        v2f B0 = bp[0];
        v2f B1 = bp[16];
        v2f B2 = bp[32];
        v2f B3 = bp[48];
        acc0 = __builtin_amdgcn_wmma_f32_16x16x4_f32(false, A, false, B0, (short)0, acc0, false, false);
        acc1 = __builtin_amdgcn_wmma_f32_16x16x4_f32(false, A, false, B1, (short)0, acc1, false, false);
        acc2 = __builtin_amdgcn_wmma_f32_16x16x4_f32(false, A, false, B2, (short)0, acc2, false, false);
        acc3 = __builtin_amdgcn_wmma_f32_16x16x4_f32(false, A, false, B3, (short)0, acc3, false, false);
    }

    // D layout: VGPR v -> row (tile*16 + v + 8*hh), col = nt*16 + lr
    float* __restrict__ outp = hw1 + (long)(tile * 16 + 8 * hh) * HID + lr;
#pragma unroll
    for (int v = 0; v < 8; ++v) {
        long o = (long)v * HID;
        outp[o +  0] = acc0[v];
        outp[o + 16] = acc1[v];
        outp[o + 32] = acc2[v];
        outp[o + 48] = acc3[v];
    }
}

// ---------------- edge scatter, layer 1: agg[dst] += hw1[src] * en ----------------
// 16 threads per edge, float4 per thread.
__global__ void scatter1(const int* __restrict__ src, const int* __restrict__ dst,
                         const float* __restrict__ en, const float* __restrict__ hw1,
                         float* __restrict__ agg) {
    int tid = blockIdx.x * blockDim.x + threadIdx.x;  // E*16 threads
    int e = tid >> 4;
    int f = (tid & 15) * 4;
    if (e >= N_EDGES) return;
    int s = src[e], d = dst[e];
    float w = en[e];
    v4f m = *(const v4f*)(hw1 + (long)s * HID + f);
    float* ap = agg + (long)d * HID + f;
    unsafeAtomicAdd(ap + 0, m.x * w);
    unsafeAtomicAdd(ap + 1, m.y * w);
    unsafeAtomicAdd(ap + 2, m.z * w);
    unsafeAtomicAdd(ap + 3, m.w * w);
}

// ---------------- h = relu(agg + hw1*self_norm + b1), in-place on agg ----------------
__global__ void relu_bias(const float* __restrict__ hw1, const float* __restrict__ dinv,
                          const float* __restrict__ b1, float* __restrict__ h) {
    int tid = blockIdx.x * blockDim.x + threadIdx.x;
    if (tid >= N_NODES * HID) return;
    int n = tid >> 6, f = tid & 63;
    float sn = dinv[n]; sn *= sn;
    float v = h[tid] + hw1[tid] * sn + b1[f];
    h[tid] = v > 0.0f ? v : 0.0f;
}

// ---------------- GEMM2: hw2[N,2] = h[N,64] @ W2[64,2] ----------------
__global__ void gemm2(const float* __restrict__ h, const float* __restrict__ W2,
                      float* __restrict__ hw2) {
    int n = blockIdx.x * blockDim.x + threadIdx.x;
    if (n >= N_NODES) return;
    const float* hr = h + (long)n * HID;
    float o0 = 0.0f, o1 = 0.0f;
#pragma unroll 8
    for (int k = 0; k < HID; ++k) {
        float hv = hr[k];
        o0 = fmaf(hv, W2[2 * k],     o0);
        o1 = fmaf(hv, W2[2 * k + 1], o1);
    }
    hw2[2 * n]     = o0;
    hw2[2 * n + 1] = o1;
}

// ---------------- edge scatter, layer 2: out[dst] += hw2[src] * en ----------------
__global__ void scatter2(const int* __restrict__ src, const int* __restrict__ dst,
                         const float* __restrict__ en, const float* __restrict__ hw2,
                         float* __restrict__ out) {
    int e = blockIdx.x * blockDim.x + threadIdx.x;
    if (e >= N_EDGES) return;
    int s = src[e], d = dst[e];
    float w = en[e];
    unsafeAtomicAdd(&out[2 * d],     hw2[2 * s]     * w);
    unsafeAtomicAdd(&out[2 * d + 1], hw2[2 * s + 1] * w);
}

// ---------------- out += hw2*self_norm + b2 ----------------
__global__ void finish2(const float* __restrict__ hw2, const float* __restrict__ dinv,
                        const float* __restrict__ b2, float* __restrict__ out) {
    int n = blockIdx.x * blockDim.x + threadIdx.x;
    if (n >= N_NODES) return;
    float sn = dinv[n]; sn *= sn;
    out[2 * n]     += hw2[2 * n]     * sn + b2[0];
    out[2 * n + 1] += hw2[2 * n + 1] * sn + b2[1];
}

extern "C" void kernel_launch(void* const* d_in, const int* in_sizes, int n_in,
                              void* d_out, int out_size, void* d_ws, size_t ws_size,
                              hipStream_t stream) {
    const float* x  = (const float*)d_in[0];
    const int*   ei = (const int*)  d_in[1];   // [2, E]
    const float* W1 = (const float*)d_in[2];
    const float* b1 = (const float*)d_in[3];
    const float* W2 = (const float*)d_in[4];
    const float* b2 = (const float*)d_in[5];
    const int* src = ei;
    const int* dst = ei + N_EDGES;
    float* out = (float*)d_out;

    // workspace layout (floats): dinv[N] | en[E] | hw1[64N] | agg1/h[64N] | hw2[2N]
    float* ws   = (float*)d_ws;
    float* dinv = ws;
    float* en   = dinv + N_NODES;
    float* hw1  = en + N_EDGES;
    float* h    = hw1 + (size_t)N_NODES * HID;
    float* hw2  = h   + (size_t)N_NODES * HID;

    const dim3 blk(256);

    // zero accumulators (deg buffer, agg1 buffer, output)
    zero_kernel<<<(N_NODES + 255) / 256, blk, 0, stream>>>(dinv, N_NODES);
    zero_kernel<<<(N_NODES * HID + 255) / 256, blk, 0, stream>>>(h, N_NODES * HID);
    zero_kernel<<<(OUT_DIM * N_NODES + 255) / 256, blk, 0, stream>>>(out, OUT_DIM * N_NODES);

    // normalization coefficients
    deg_kernel  <<<(N_EDGES + 255) / 256, blk, 0, stream>>>(dst, dinv);
    dinv_kernel <<<(N_NODES + 255) / 256, blk, 0, stream>>>(dinv);
    enorm_kernel<<<(N_EDGES + 255) / 256, blk, 0, stream>>>(src, dst, dinv, en);

    // layer 1
    gemm1_wmma<<<(NTILES + 7) / 8, blk, 0, stream>>>(x, W1, hw1);
    scatter1  <<<(N_EDGES * 16) / 256, blk, 0, stream>>>(src, dst, en, hw1, h);
    relu_bias <<<(N_NODES * HID + 255) / 256, blk, 0, stream>>>(hw1, dinv, b1, h);

    // layer 2
    gemm2   <<<(N_NODES + 255) / 256, blk, 0, stream>>>(h, W2, hw2);
    scatter2<<<(N_EDGES + 255) / 256, blk, 0, stream>>>(src, dst, en, hw2, out);
    finish2 <<<(N_NODES + 255) / 256, blk, 0, stream>>>(hw2, dinv, b2, out);
}